// RWKV_RNN_86655260164113
// MI455X (gfx1250) — compile-verified
//
#include <hip/hip_runtime.h>
#include <hip/hip_bf16.h>

#define LL   24
#define CC   1024
#define HH   16
#define SS   64
#define VV   50257
#define FFF  3584
#define EPSF 1e-5f

typedef __attribute__((ext_vector_type(2))) float v2f;
typedef __attribute__((ext_vector_type(8))) float v8f;

// ---------------- reduction helper (256-thread block) ----------------
__device__ __forceinline__ float block_sum256(float v, float* sm) {
    int t = threadIdx.x;
    sm[t] = v; __syncthreads();
    for (int o = 128; o > 0; o >>= 1) {
        if (t < o) sm[t] += sm[t + o];
        __syncthreads();
    }
    float r = sm[0];
    __syncthreads();
    return r;
}

// ---------------- x = layernorm(emb[token]) ----------------
__global__ __launch_bounds__(256)
void k_embed(const int* __restrict__ tok, const float* __restrict__ emb,
             const float* __restrict__ w, const float* __restrict__ b,
             float* __restrict__ x) {
    __shared__ float sm[256];
    int t = threadIdx.x;
    const float* e = emb + (size_t)tok[0] * CC;
    float s = 0.f;
    for (int i = t; i < CC; i += 256) s += e[i];
    float mu = block_sum256(s, sm) * (1.f / CC);
    float vs = 0.f;
    for (int i = t; i < CC; i += 256) { float d = e[i] - mu; vs += d * d; }
    float inv = rsqrtf(block_sum256(vs, sm) * (1.f / CC) + EPSF);
    for (int i = t; i < CC; i += 256) x[i] = (e[i] - mu) * inv * w[i] + b[i];
}

// ---------------- generic layernorm: out = ln(in)*w+b ----------------
__global__ __launch_bounds__(256)
void k_ln(const float* __restrict__ in, const float* __restrict__ w,
          const float* __restrict__ b, float* __restrict__ out) {
    __shared__ float sm[256];
    int t = threadIdx.x;
    float s = 0.f;
    for (int i = t; i < CC; i += 256) s += in[i];
    float mu = block_sum256(s, sm) * (1.f / CC);
    float vs = 0.f;
    for (int i = t; i < CC; i += 256) { float d = in[i] - mu; vs += d * d; }
    float inv = rsqrtf(block_sum256(vs, sm) * (1.f / CC) + EPSF);
    for (int i = t; i < CC; i += 256) out[i] = (in[i] - mu) * inv * w[i] + b[i];
}

// ---------------- time-mix prologue: ln1 + token-shift mixes ----------------
__global__ __launch_bounds__(256)
void k_time_mix_pre(const float* __restrict__ x, const float* __restrict__ w,
                    const float* __restrict__ b,
                    const float* __restrict__ tmk, const float* __restrict__ tmv,
                    const float* __restrict__ tmr, const float* __restrict__ prev,
                    float* __restrict__ xk, float* __restrict__ xv,
                    float* __restrict__ xr, float* __restrict__ xx_out) {
    __shared__ float sm[256];
    int t = threadIdx.x;
    float s = 0.f;
    for (int i = t; i < CC; i += 256) s += x[i];
    float mu = block_sum256(s, sm) * (1.f / CC);
    float vs = 0.f;
    for (int i = t; i < CC; i += 256) { float d = x[i] - mu; vs += d * d; }
    float inv = rsqrtf(block_sum256(vs, sm) * (1.f / CC) + EPSF);
    for (int i = t; i < CC; i += 256) {
        float xx = (x[i] - mu) * inv * w[i] + b[i];
        float p = prev[i];
        xx_out[i] = xx;
        float a = tmk[i]; xk[i] = xx * a + p * (1.f - a);
        float c = tmv[i]; xv[i] = xx * c + p * (1.f - c);
        float r = tmr[i]; xr[i] = xx * r + p * (1.f - r);
    }
}

// ---------------- channel-mix prologue: ln2 + mixes ----------------
__global__ __launch_bounds__(256)
void k_chan_pre(const float* __restrict__ x, const float* __restrict__ w,
                const float* __restrict__ b,
                const float* __restrict__ tmk, const float* __restrict__ tmr,
                const float* __restrict__ prev,
                float* __restrict__ fk, float* __restrict__ fr,
                float* __restrict__ xx_out) {
    __shared__ float sm[256];
    int t = threadIdx.x;
    float s = 0.f;
    for (int i = t; i < CC; i += 256) s += x[i];
    float mu = block_sum256(s, sm) * (1.f / CC);
    float vs = 0.f;
    for (int i = t; i < CC; i += 256) { float d = x[i] - mu; vs += d * d; }
    float inv = rsqrtf(block_sum256(vs, sm) * (1.f / CC) + EPSF);
    for (int i = t; i < CC; i += 256) {
        float xx = (x[i] - mu) * inv * w[i] + b[i];
        float p = prev[i];
        xx_out[i] = xx;
        float a = tmk[i]; fk[i] = xx * a + p * (1.f - a);
        float r = tmr[i]; fr[i] = xx * r + p * (1.f - r);
    }
}

// ---------------- WMMA-based GEMV: y[0..M) (+)= op(W[MxK] @ x) ----------------
// One wave computes a 16-row tile via V_WMMA_F32_16X16X4_F32.
// A layout (ISA 7.12.2): lanes 0-15 hold {K=0,K=1}, lanes 16-31 hold {K=2,K=3}
// of rows M=lane&15 -> a single aligned 8B load per lane per k-step.
// B is x broadcast across all 16 columns, so every column of D holds the same
// 16 dot products; lane 0 / lane 16 each extract 8 of them from the 8 acc VGPRs.
// modes: 0=store  1=y+=d  2=sigmoid  3=relu(d)^2  4=y+=gate*d
__global__ __launch_bounds__(256)
void k_gemv(const float* __restrict__ W, const float* __restrict__ x,
            float* __restrict__ y, const float* __restrict__ gate,
            int M, int K, int mode) {
    int wave = (int)((blockIdx.x * blockDim.x + threadIdx.x) >> 5);
    int lane = threadIdx.x & 31;
    int row0 = wave * 16;
    if (row0 >= M) return;                 // wave-uniform exit: EXEC stays all-1s at WMMA
    int half = lane >> 4;
    int mrow = lane & 15;
    const float* wrow = W + (size_t)(row0 + mrow) * K + 2 * half;
    const float* xp   = x + 2 * half;
    v8f acc = {};
    #pragma unroll 8
    for (int k = 0; k < K; k += 4) {
        v2f a  = *(const v2f*)(wrow + k);
        v2f bb = *(const v2f*)(xp + k);
        acc = __builtin_amdgcn_wmma_f32_16x16x4_f32(
                  false, a, false, bb, (short)0, acc, false, false);
    }
    if (mrow == 0) {                       // lane 0 -> rows row0..+7, lane 16 -> rows +8..+15
        int rbase = row0 + 8 * half;
        #pragma unroll
        for (int i = 0; i < 8; ++i) {
            int r = rbase + i;
            if (r < M) {
                float d = acc[i];
                if (mode == 0)      y[r] = d;
                else if (mode == 1) y[r] += d;
                else if (mode == 2) y[r] = 1.f / (1.f + expf(-d));
                else if (mode == 3) { float t = d > 0.f ? d : 0.f; y[r] = t * t; }
                else                y[r] += gate[r] * d;
            }
        }
    }
}

// ---------------- WKV step, one head per block ----------------
// out_h = tf*(r.k)*v + r@s ;  new_s = k v^T + td*s   (s read from input state,
// written to output state; inputs never mutated)
__global__ __launch_bounds__(256)
void k_wkv(const float* __restrict__ r, const float* __restrict__ k,
           const float* __restrict__ v, const float* __restrict__ s_in,
           float* __restrict__ s_out, const float* __restrict__ tf,
           const float* __restrict__ td, float* __restrict__ out) {
    __shared__ float rs[SS], ks[SS], vs[SS], part[4][SS];
    __shared__ float rk;
    int h = blockIdx.x, t = threadIdx.x;
    if (t < SS) { rs[t] = r[h * SS + t]; ks[t] = k[h * SS + t]; vs[t] = v[h * SS + t]; }
    __syncthreads();
    if (t == 0) {
        float a = 0.f;
        for (int i = 0; i < SS; ++i) a += rs[i] * ks[i];
        rk = a;
    }
    int j = t & 63, ii = t >> 6;
    const float* sb_in = s_in + (size_t)h * SS * SS;
    float*       sb_ot = s_out + (size_t)h * SS * SS;
    float tdv = td[h], tfv = tf[h];
    float sum = 0.f;
    #pragma unroll 4
    for (int i = ii * 16; i < ii * 16 + 16; ++i) {
        float so = sb_in[i * SS + j];
        sum += rs[i] * so;
        sb_ot[i * SS + j] = ks[i] * vs[j] + tdv * so;
    }
    part[ii][j] = sum;
    __syncthreads();
    if (t < SS)
        out[h * SS + t] = part[0][t] + part[1][t] + part[2][t] + part[3][t]
                        + tfv * rk * vs[t];
}

// ---------------- group norm (H groups of S), one head per block ----------------
__global__ __launch_bounds__(64)
void k_groupnorm(const float* __restrict__ xin, const float* __restrict__ w,
                 const float* __restrict__ b, float* __restrict__ gn) {
    __shared__ float sm[SS];
    int h = blockIdx.x, t = threadIdx.x;
    float v = xin[h * SS + t];
    sm[t] = v; __syncthreads();
    for (int o = 32; o > 0; o >>= 1) { if (t < o) sm[t] += sm[t + o]; __syncthreads(); }
    float mu = sm[0] * (1.f / SS); __syncthreads();
    float d = v - mu;
    sm[t] = d * d; __syncthreads();
    for (int o = 32; o > 0; o >>= 1) { if (t < o) sm[t] += sm[t + o]; __syncthreads(); }
    float inv = rsqrtf(sm[0] * (1.f / SS) + EPSF);
    int c = h * SS + t;
    gn[c] = d * inv * w[c] + b[c];
}

// ---------------- host orchestration ----------------
extern "C" void kernel_launch(void* const* d_in, const int* in_sizes, int n_in,
                              void* d_out, int out_size, void* d_ws, size_t ws_size,
                              hipStream_t stream) {
    const int*   token   = (const int*)  d_in[0];
    const float* state   = (const float*)d_in[1];
    const float* emb     = (const float*)d_in[2];
    const float* ln0_w   = (const float*)d_in[3];
    const float* ln0_b   = (const float*)d_in[4];
    const float* ln1_w   = (const float*)d_in[5];
    const float* ln1_b   = (const float*)d_in[6];
    const float* ln2_w   = (const float*)d_in[7];
    const float* ln2_b   = (const float*)d_in[8];
    const float* att_tmk = (const float*)d_in[9];
    const float* att_tmv = (const float*)d_in[10];
    const float* att_tmr = (const float*)d_in[11];
    const float* att_tf  = (const float*)d_in[12];
    const float* att_td  = (const float*)d_in[13];
    const float* att_kw  = (const float*)d_in[14];
    const float* att_vw  = (const float*)d_in[15];
    const float* att_rw  = (const float*)d_in[16];
    const float* att_ow  = (const float*)d_in[17];
    const float* lnx_w   = (const float*)d_in[18];
    const float* lnx_b   = (const float*)d_in[19];
    const float* ffn_tmk = (const float*)d_in[20];
    const float* ffn_tmr = (const float*)d_in[21];
    const float* ffn_kw  = (const float*)d_in[22];
    const float* ffn_vw  = (const float*)d_in[23];
    const float* ffn_rw  = (const float*)d_in[24];
    const float* lnout_w = (const float*)d_in[25];
    const float* lnout_b = (const float*)d_in[26];
    const float* head_w  = (const float*)d_in[27];

    float* logits = (float*)d_out;          // [V]
    float* st_out = (float*)d_out + VV;     // [L*(2+S)*C]

    float* ws  = (float*)d_ws;
    float* x   = ws + 0 * CC;
    float* xk  = ws + 1 * CC;
    float* xv  = ws + 2 * CC;
    float* xr  = ws + 3 * CC;
    float* rb  = ws + 4 * CC;
    float* kb  = ws + 5 * CC;
    float* vb  = ws + 6 * CC;
    float* ob  = ws + 7 * CC;
    float* gn  = ws + 8 * CC;
    float* fk  = ws + 9 * CC;
    float* fr  = ws + 10 * CC;
    float* rrb = ws + 11 * CC;
    float* xf  = ws + 12 * CC;
    float* kkb = ws + 13 * CC;              // [FF]

    k_embed<<<1, 256, 0, stream>>>(token, emb, ln0_w, ln0_b, x);

    for (int l = 0; l < LL; ++l) {
        const float* bin  = state  + (size_t)l * (2 + SS) * CC;
        float*       bout = st_out + (size_t)l * (2 + SS) * CC;
        const size_t wl   = (size_t)l * CC * CC;

        // --- time mixing ---
        k_time_mix_pre<<<1, 256, 0, stream>>>(x, ln1_w + l * CC, ln1_b + l * CC,
            att_tmk + l * CC, att_tmv + l * CC, att_tmr + l * CC,
            bin + CC, xk, xv, xr, bout + CC);
        k_gemv<<<8, 256, 0, stream>>>(att_rw + wl, xr, rb, nullptr, CC, CC, 0);
        k_gemv<<<8, 256, 0, stream>>>(att_kw + wl, xk, kb, nullptr, CC, CC, 0);
        k_gemv<<<8, 256, 0, stream>>>(att_vw + wl, xv, vb, nullptr, CC, CC, 0);
        k_wkv<<<HH, 256, 0, stream>>>(rb, kb, vb, bin + 2 * CC, bout + 2 * CC,
                                      att_tf + l * HH, att_td + l * HH, ob);
        k_groupnorm<<<HH, 64, 0, stream>>>(ob, lnx_w + l * CC, lnx_b + l * CC, gn);
        k_gemv<<<8, 256, 0, stream>>>(att_ow + wl, gn, x, nullptr, CC, CC, 1); // x += ow@gn

        // --- channel mixing ---
        k_chan_pre<<<1, 256, 0, stream>>>(x, ln2_w + l * CC, ln2_b + l * CC,
            ffn_tmk + l * CC, ffn_tmr + l * CC, bin, fk, fr, bout);
        k_gemv<<<8, 256, 0, stream>>>(ffn_rw + wl, fr, rrb, nullptr, CC, CC, 2); // sigmoid
        k_gemv<<<28, 256, 0, stream>>>(ffn_kw + (size_t)l * FFF * CC, fk, kkb,
                                       nullptr, FFF, CC, 3);                     // relu^2
        k_gemv<<<8, 256, 0, stream>>>(ffn_vw + (size_t)l * CC * FFF, kkb, x,
                                      rrb, CC, FFF, 4);                          // x += rr*(fvw@kk)
    }

    // --- head ---
    k_ln<<<1, 256, 0, stream>>>(x, lnout_w, lnout_b, xf);
    // ceil(V/16)=3142 wave-tiles, 8 waves/block -> 393 blocks (tail waves exit)
    k_gemv<<<393, 256, 0, stream>>>(head_w, xf, logits, nullptr, VV, CC, 0);
}